// SpatialAttentionModule_50242527429217
// MI455X (gfx1250) — compile-verified
//
#include <hip/hip_runtime.h>
#include <math.h>

typedef float v2f __attribute__((ext_vector_type(2)));
typedef float v8f __attribute__((ext_vector_type(8)));

#define BDIM 256

// ---------------------------------------------------------------------------
// zero the 12 scalar accumulators: [0..8] = a-parts (stage*3+q), [9..11] = bl
// ---------------------------------------------------------------------------
__global__ void init_accum_kernel(float* accum) {
  if (threadIdx.x < 12) accum[threadIdx.x] = 0.0f;
}

// ---------------------------------------------------------------------------
// adaptive 4x4 average pool: x [B*C, H, H] -> out [B*C, 16]
// one block per (b,c); 16 threads per output bin
// ---------------------------------------------------------------------------
template <int H>
__global__ void pool4_kernel(const float* __restrict__ x, float* __restrict__ out) {
  __shared__ float sh[BDIM];
  const int tid = threadIdx.x;
  const int bin = tid >> 4;      // 0..15
  const int j   = tid & 15;      // lane within bin group
  const int by = bin >> 2, bx = bin & 3;
  const size_t base = (size_t)blockIdx.x * H * H;
  const int h0 = (by * H) >> 2, h1 = ((by + 1) * H) >> 2;
  const int w0 = (bx * H) >> 2, w1 = ((bx + 1) * H) >> 2;
  const int bw = w1 - w0;
  const int n  = (h1 - h0) * bw;
  float sum = 0.f;
  for (int e = j; e < n; e += 16) {
    int hh = h0 + e / bw;
    int ww = w0 + e % bw;
    sum += x[base + hh * H + ww];
  }
  sh[tid] = sum;
  __syncthreads();
#pragma unroll
  for (int s = 8; s > 0; s >>= 1) {
    if (j < s) sh[tid] += sh[tid + s];
    __syncthreads();
  }
  if (j == 0) out[(size_t)blockIdx.x * 16 + bin] = sh[tid] / (float)n;
}

// ---------------------------------------------------------------------------
// per-batch relation kernel: builds rel4/rel2/rel1 (t & s) from the 4x4 pool,
// column-normalizes, accumulates Frobenius^2 diffs into aParts[0..2], and
// writes re_t / re_s = pool4 + alpha*(w4 + rep4(w2) + rep16(w1)).
// grid = B, block = 256 (== 16x16 Gram entries)
// ---------------------------------------------------------------------------
template <int C, int H>
__global__ void relation_kernel(const float* __restrict__ poolT,
                                const float* __restrict__ poolS,
                                float* __restrict__ reT,
                                float* __restrict__ reS,
                                const float* __restrict__ alphaP,
                                float* __restrict__ aParts) {
  __shared__ float u4t[256], u4s[256];
  __shared__ float u2t[16],  u2s[16];
  __shared__ float n4t[16],  n4s[16];
  __shared__ float n2t[4],   n2s[4];
  __shared__ float r1t, r1s;
  __shared__ float red[BDIM];
  __shared__ float wq2[16], wq1[16];   // adaptive-bin weights (4x4 -> 2x2 / 1x1)

  const int tid = threadIdx.x;
  const float alpha = alphaP[0];
  const size_t bbase = (size_t)blockIdx.x * C * 16;

  if (tid < 16) {
    int by = tid >> 2, bx = tid & 3;
    int cy = (((by + 1) * H) >> 2) - ((by * H) >> 2);
    int cx = (((bx + 1) * H) >> 2) - ((bx * H) >> 2);
    float c4 = (float)(cy * cx);
    int gy0 = (by >> 1) << 1, gx0 = (bx >> 1) << 1;
    int gcy = (((gy0 + 2) * H) >> 2) - ((gy0 * H) >> 2);
    int gcx = (((gx0 + 2) * H) >> 2) - ((gx0 * H) >> 2);
    wq2[tid] = c4 / (float)(gcy * gcx);
    wq1[tid] = c4 / (float)(H * H);
  }
  __syncthreads();

  // phase 1: unnormalized 16x16 Grams (sum over channels)
  {
    const int i = tid >> 4, j = tid & 15;
    float st = 0.f, ss = 0.f;
    for (int c = 0; c < C; ++c) {
      const float* pt = poolT + bbase + c * 16;
      const float* ps = poolS + bbase + c * 16;
      st += pt[i] * pt[j];
      ss += ps[i] * ps[j];
    }
    u4t[tid] = st;
    u4s[tid] = ss;
  }
  __syncthreads();

  // phase 2: q=2 Gram via weighted group sums of u4; column norms of u4; q=1
  if (tid < 16) {
    const int i2 = tid >> 2, j2 = tid & 3;
    const int Y = i2 >> 1, X = i2 & 1, Y2 = j2 >> 1, X2 = j2 & 1;
    float st = 0.f, ss = 0.f;
#pragma unroll
    for (int dy = 0; dy < 2; ++dy)
#pragma unroll
      for (int dx = 0; dx < 2; ++dx) {
        int i = (2 * Y + dy) * 4 + (2 * X + dx);
#pragma unroll
        for (int ey = 0; ey < 2; ++ey)
#pragma unroll
          for (int ex = 0; ex < 2; ++ex) {
            int jj = (2 * Y2 + ey) * 4 + (2 * X2 + ex);
            float w = wq2[i] * wq2[jj];
            st += w * u4t[i * 16 + jj];
            ss += w * u4s[i * 16 + jj];
          }
      }
    u2t[tid] = st;
    u2s[tid] = ss;
    float nt = 0.f, ns = 0.f;
    for (int i = 0; i < 16; ++i) {
      float a = u4t[i * 16 + tid]; nt += a * a;
      float b = u4s[i * 16 + tid]; ns += b * b;
    }
    n4t[tid] = sqrtf(nt);
    n4s[tid] = sqrtf(ns);
  } else if (tid == 32) {
    float st = 0.f, ss = 0.f;
    for (int i = 0; i < 16; ++i)
      for (int jj = 0; jj < 16; ++jj) {
        float w = wq1[i] * wq1[jj];
        st += w * u4t[i * 16 + jj];
        ss += w * u4s[i * 16 + jj];
      }
    r1t = st / fmaxf(fabsf(st), 1e-12f);
    r1s = ss / fmaxf(fabsf(ss), 1e-12f);
  }
  __syncthreads();

  // phase 3: q=2 column norms
  if (tid < 4) {
    float nt = 0.f, ns = 0.f;
    for (int i2 = 0; i2 < 4; ++i2) {
      float a = u2t[i2 * 4 + tid]; nt += a * a;
      float b = u2s[i2 * 4 + tid]; ns += b * b;
    }
    n2t[tid] = sqrtf(nt);
    n2s[tid] = sqrtf(ns);
  }
  __syncthreads();

  // phase 4: normalize in place
  {
    int j = tid & 15;
    u4t[tid] /= fmaxf(n4t[j], 1e-12f);
    u4s[tid] /= fmaxf(n4s[j], 1e-12f);
    if (tid < 16) {
      int j2 = tid & 3;
      u2t[tid] /= fmaxf(n2t[j2], 1e-12f);
      u2s[tid] /= fmaxf(n2s[j2], 1e-12f);
    }
  }
  __syncthreads();

  // phase 5: accumulate sum((rel_t - rel_s)^2) per q across batches
  {
    float d = u4t[tid] - u4s[tid];
    red[tid] = d * d;
  }
  __syncthreads();
  for (int s = BDIM / 2; s > 0; s >>= 1) {
    if (tid < s) red[tid] += red[tid + s];
    __syncthreads();
  }
  if (tid == 0) {
    atomicAdd(&aParts[0], red[0]);
    float d2 = 0.f;
    for (int i = 0; i < 16; ++i) { float d = u2t[i] - u2s[i]; d2 += d * d; }
    atomicAdd(&aParts[1], d2);
    float d1 = r1t - r1s;
    atomicAdd(&aParts[2], d1 * d1);
  }
  __syncthreads();

  // phase 6: re = rf4 + alpha*(rf4@rel4n + rep4(rf2@rel2n) + rf1*rel1n)
  for (int c = tid; c < C; c += BDIM) {
    float rt4[16], rs4[16];
    const float* pt = poolT + bbase + c * 16;
    const float* ps = poolS + bbase + c * 16;
#pragma unroll
    for (int i = 0; i < 16; ++i) { rt4[i] = pt[i]; rs4[i] = ps[i]; }
    float rt2[4] = {0.f, 0.f, 0.f, 0.f}, rs2[4] = {0.f, 0.f, 0.f, 0.f};
    float rt1 = 0.f, rs1 = 0.f;
#pragma unroll
    for (int i = 0; i < 16; ++i) {
      int by = i >> 2, bx = i & 3;
      int i2 = ((by >> 1) << 1) + (bx >> 1);
      rt2[i2] += wq2[i] * rt4[i];
      rs2[i2] += wq2[i] * rs4[i];
      rt1 += wq1[i] * rt4[i];
      rs1 += wq1[i] * rs4[i];
    }
    float* ot = reT + bbase + c * 16;
    float* os = reS + bbase + c * 16;
#pragma unroll
    for (int k = 0; k < 16; ++k) {
      float wt = 0.f, wsv = 0.f;
#pragma unroll
      for (int i = 0; i < 16; ++i) {
        wt  += rt4[i] * u4t[i * 16 + k];
        wsv += rs4[i] * u4s[i * 16 + k];
      }
      const int k2 = k >> 2;
#pragma unroll
      for (int i2 = 0; i2 < 4; ++i2) {
        wt  += rt2[i2] * u2t[i2 * 4 + k2];
        wsv += rs2[i2] * u2s[i2 * 4 + k2];
      }
      wt  += rt1 * r1t;
      wsv += rs1 * r1s;
      ot[k] = rt4[k] + alpha * wt;
      os[k] = rs4[k] + alpha * wsv;
    }
  }
}

// ---------------------------------------------------------------------------
// fused Linear + ||ft - fs||^2 partial sum. fs = reS @ W^T + bias; ft = reT.
// GEMM: M=32 (2 WMMA row tiles), N = 16 cols per block, contiguous split-K
// chunks across 8 waves; double-buffered fragment loads overlap HBM latency
// with v_wmma_f32_16x16x4_f32; wave partials reduced in LDS via ds_add_f32.
// grid = K/16, block = 256.
// ---------------------------------------------------------------------------
template <int K>
__global__ void linear_diff_kernel(const float* __restrict__ A,    // reS [32,K]
                                   const float* __restrict__ Ft,   // reT [32,K]
                                   const float* __restrict__ Wm,   // [K,K] (out,in)
                                   const float* __restrict__ bias, // [K]
                                   float* __restrict__ blPart) {
  __shared__ float fsbuf[512];   // fs tile [32 rows][16 cols]
  __shared__ float red[BDIM];
  const int tid  = threadIdx.x;
  const int lane = tid & 31;
  // wave id as a scalar -> uniform loop bounds, no EXEC masking around WMMA
  const int wv   = __builtin_amdgcn_readfirstlane(tid >> 5);
  const int o0   = blockIdx.x * 16;
  const int row  = lane & 15;           // M index within tile / N index for B
  const int col  = o0 + row;            // output feature handled by this lane
  const int khalf = (lane < 16) ? 0 : 2;

  for (int i = tid; i < 512; i += BDIM) fsbuf[i] = 0.f;
  __syncthreads();

  v8f acc0 = {0.f, 0.f, 0.f, 0.f, 0.f, 0.f, 0.f, 0.f};
  v8f acc1 = {0.f, 0.f, 0.f, 0.f, 0.f, 0.f, 0.f, 0.f};

  constexpr int KC = K >> 3;            // contiguous K-chunk per wave
  const int kbeg = wv * KC;
  const int kend = kbeg + KC;
  const float* wp  = Wm + (size_t)col * K + khalf;
  const float* ap0 = A + (size_t)row * K + khalf;
  const float* ap1 = A + (size_t)(row + 16) * K + khalf;

  // software pipeline: prologue loads
  v2f bw = *(const v2f*)(wp + kbeg);
  v2f a0 = *(const v2f*)(ap0 + kbeg);
  v2f a1 = *(const v2f*)(ap1 + kbeg);

  for (int k = kbeg + 4; k < kend; k += 4) {
    v2f bwN = *(const v2f*)(wp + k);
    v2f a0N = *(const v2f*)(ap0 + k);
    v2f a1N = *(const v2f*)(ap1 + k);
    if ((k & 31) == 0)                       // once per 128B line, 512B ahead
      __builtin_prefetch(wp + k + 128, 0, 3);
    acc0 = __builtin_amdgcn_wmma_f32_16x16x4_f32(false, a0, false, bw,
                                                 (short)0, acc0, false, false);
    acc1 = __builtin_amdgcn_wmma_f32_16x16x4_f32(false, a1, false, bw,
                                                 (short)0, acc1, false, false);
    bw = bwN; a0 = a0N; a1 = a1N;
  }
  // epilogue
  acc0 = __builtin_amdgcn_wmma_f32_16x16x4_f32(false, a0, false, bw,
                                               (short)0, acc0, false, false);
  acc1 = __builtin_amdgcn_wmma_f32_16x16x4_f32(false, a1, false, bw,
                                               (short)0, acc1, false, false);

  // D layout: lane l, vgpr v -> M = v + (l<16?0:8), N = l&15
  const int nbase = (lane < 16) ? 0 : 8;
#pragma unroll
  for (int v = 0; v < 8; ++v) {
    atomicAdd(&fsbuf[(nbase + v) * 16 + row],      acc0[v]);   // ds_add_f32
    atomicAdd(&fsbuf[(16 + nbase + v) * 16 + row], acc1[v]);
  }
  __syncthreads();

  float local = 0.f;
  for (int i = tid; i < 512; i += BDIM) {
    int n = i >> 4;
    int o = o0 + (i & 15);
    float fs = fsbuf[i] + bias[o];
    float d = Ft[(size_t)n * K + o] - fs;
    local += d * d;
  }
  red[tid] = local;
  __syncthreads();
  for (int s = BDIM / 2; s > 0; s >>= 1) {
    if (tid < s) red[tid] += red[tid + s];
    __syncthreads();
  }
  if (tid == 0) atomicAdd(blPart, red[0]);
}

// ---------------------------------------------------------------------------
// final: a = sum sqrt(a-parts); bl = sum sqrt(bl-parts)
// ---------------------------------------------------------------------------
__global__ void finalize_kernel(const float* __restrict__ accum, float* __restrict__ out) {
  if (threadIdx.x == 0 && blockIdx.x == 0) {
    float a = 0.f;
    for (int i = 0; i < 9; ++i) a += sqrtf(accum[i]);
    float bl = 0.f;
    for (int i = 0; i < 3; ++i) bl += sqrtf(accum[9 + i]);
    out[0] = a;
    out[1] = bl;
  }
}

extern "C" void kernel_launch(void* const* d_in, const int* in_sizes, int n_in,
                              void* d_out, int out_size, void* d_ws, size_t ws_size,
                              hipStream_t stream) {
  (void)in_sizes; (void)n_in; (void)out_size; (void)ws_size;

  const float* t[3] = {(const float*)d_in[0], (const float*)d_in[1], (const float*)d_in[2]};
  const float* s[3] = {(const float*)d_in[3], (const float*)d_in[4], (const float*)d_in[5]};
  const float* alpha = (const float*)d_in[6];
  const float* Wm[3]   = {(const float*)d_in[7], (const float*)d_in[9],  (const float*)d_in[11]};
  const float* bias[3] = {(const float*)d_in[8], (const float*)d_in[10], (const float*)d_in[12]};

  const int Bv = 32;
  const int Cs[3] = {128, 256, 512};

  float* ws = (float*)d_ws;
  size_t off = 0;
  float *pt[3], *ps[3], *rt[3], *rs[3];
  for (int i = 0; i < 3; ++i) { pt[i] = ws + off; off += (size_t)Bv * Cs[i] * 16; }
  for (int i = 0; i < 3; ++i) { ps[i] = ws + off; off += (size_t)Bv * Cs[i] * 16; }
  for (int i = 0; i < 3; ++i) { rt[i] = ws + off; off += (size_t)Bv * Cs[i] * 16; }
  for (int i = 0; i < 3; ++i) { rs[i] = ws + off; off += (size_t)Bv * Cs[i] * 16; }
  float* accum = ws + off;

  init_accum_kernel<<<1, 32, 0, stream>>>(accum);

  pool4_kernel<56><<<Bv * Cs[0], BDIM, 0, stream>>>(t[0], pt[0]);
  pool4_kernel<56><<<Bv * Cs[0], BDIM, 0, stream>>>(s[0], ps[0]);
  pool4_kernel<28><<<Bv * Cs[1], BDIM, 0, stream>>>(t[1], pt[1]);
  pool4_kernel<28><<<Bv * Cs[1], BDIM, 0, stream>>>(s[1], ps[1]);
  pool4_kernel<14><<<Bv * Cs[2], BDIM, 0, stream>>>(t[2], pt[2]);
  pool4_kernel<14><<<Bv * Cs[2], BDIM, 0, stream>>>(s[2], ps[2]);

  relation_kernel<128, 56><<<Bv, BDIM, 0, stream>>>(pt[0], ps[0], rt[0], rs[0], alpha, accum + 0);
  relation_kernel<256, 28><<<Bv, BDIM, 0, stream>>>(pt[1], ps[1], rt[1], rs[1], alpha, accum + 3);
  relation_kernel<512, 14><<<Bv, BDIM, 0, stream>>>(pt[2], ps[2], rt[2], rs[2], alpha, accum + 6);

  linear_diff_kernel<2048><<<2048 / 16, BDIM, 0, stream>>>(rs[0], rt[0], Wm[0], bias[0], accum + 9);
  linear_diff_kernel<4096><<<4096 / 16, BDIM, 0, stream>>>(rs[1], rt[1], Wm[1], bias[1], accum + 10);
  linear_diff_kernel<8192><<<8192 / 16, BDIM, 0, stream>>>(rs[2], rt[2], Wm[2], bias[2], accum + 11);

  finalize_kernel<<<1, 1, 0, stream>>>(accum, (float*)d_out);
}